// NormalizedHistogram_74637941669897
// MI455X (gfx1250) — compile-verified
//
#include <hip/hip_runtime.h>

#define NBINS 256
#define NCH 3
#define HSIZE (NBINS * NCH)   // 768 counters
#define BLOCK 256             // 8 waves of 32 on gfx1250
#define NWAVES (BLOCK / 32)
#define GRID 1536             // GRID*BLOCK*4 = 1,572,864 which is % 3 == 0 (channel phase invariance)

// ---------------------------------------------------------------------------
// Kernel 1: zero the global counter workspace (768 u32). Must run every call.
// ---------------------------------------------------------------------------
__global__ void NH_zero(unsigned int* __restrict__ ws) {
    int i = blockIdx.x * blockDim.x + threadIdx.x;
    if (i < HSIZE) ws[i] = 0u;
}

// ---------------------------------------------------------------------------
// Kernel 2: streaming histogram.
//   - float4 loads (global_load_b128), 2 loads in flight per loop iteration
//   - per-wave privatized LDS histograms (ds_add_u32), 24 KB LDS
//   - block reduction -> one global_atomic_add_u32 per bin per block
// ---------------------------------------------------------------------------
__global__ __launch_bounds__(BLOCK) void NH_hist(const float* __restrict__ in,
                                                 unsigned int* __restrict__ ws,
                                                 long long n4) {
    __shared__ unsigned int lh[NWAVES * HSIZE];
    for (int i = threadIdx.x; i < NWAVES * HSIZE; i += BLOCK) lh[i] = 0u;
    __syncthreads();

    unsigned int* h = &lh[(threadIdx.x >> 5) * HSIZE];  // wave32 private copy

    const long long tid    = (long long)blockIdx.x * BLOCK + threadIdx.x;
    const long long stride = (long long)GRID * BLOCK;
    const float4* __restrict__ in4 = (const float4*)in;

    // channel of first element handled by this thread; invariant across the
    // grid-stride loop because (stride*4) % 3 == 0.
    int c0 = (int)((tid * 4) % 3);
    int c1 = c0 + 1; if (c1 == 3) c1 = 0;
    int c2 = c1 + 1; if (c2 == 3) c2 = 0;
    const int o0 = c0 * NBINS, o1 = c1 * NBINS, o2 = c2 * NBINS;

    long long i = tid;
    // main loop: two b128 loads issued back-to-back for ILP / latency hiding
    for (; i + stride < n4; i += 2 * stride) {
        __builtin_prefetch(&in4[i + 2 * stride], 0, 0);           // global_prefetch_b8 (speculative)
        __builtin_prefetch(&in4[i + 3 * stride], 0, 0);
        float4 a = in4[i];
        float4 b = in4[i + stride];

        int a0 = min(max((int)(a.x * 256.0f), 0), 255);
        int a1 = min(max((int)(a.y * 256.0f), 0), 255);
        int a2 = min(max((int)(a.z * 256.0f), 0), 255);
        int a3 = min(max((int)(a.w * 256.0f), 0), 255);
        atomicAdd(&h[o0 + a0], 1u);
        atomicAdd(&h[o1 + a1], 1u);
        atomicAdd(&h[o2 + a2], 1u);
        atomicAdd(&h[o0 + a3], 1u);

        int b0 = min(max((int)(b.x * 256.0f), 0), 255);
        int b1 = min(max((int)(b.y * 256.0f), 0), 255);
        int b2 = min(max((int)(b.z * 256.0f), 0), 255);
        int b3 = min(max((int)(b.w * 256.0f), 0), 255);
        atomicAdd(&h[o0 + b0], 1u);
        atomicAdd(&h[o1 + b1], 1u);
        atomicAdd(&h[o2 + b2], 1u);
        atomicAdd(&h[o0 + b3], 1u);
    }
    // tail (at most one float4 per thread)
    if (i < n4) {
        float4 a = in4[i];
        int a0 = min(max((int)(a.x * 256.0f), 0), 255);
        int a1 = min(max((int)(a.y * 256.0f), 0), 255);
        int a2 = min(max((int)(a.z * 256.0f), 0), 255);
        int a3 = min(max((int)(a.w * 256.0f), 0), 255);
        atomicAdd(&h[o0 + a0], 1u);
        atomicAdd(&h[o1 + a1], 1u);
        atomicAdd(&h[o2 + a2], 1u);
        atomicAdd(&h[o0 + a3], 1u);
    }
    __syncthreads();

    // reduce the 8 wave-private copies; one global atomic per bin per block
    for (int bin = threadIdx.x; bin < HSIZE; bin += BLOCK) {
        unsigned int s = 0;
#pragma unroll
        for (int w = 0; w < NWAVES; ++w) s += lh[w * HSIZE + bin];
        if (s) atomicAdd(&ws[bin], s);
    }
}

// ---------------------------------------------------------------------------
// Kernel 3: normalize each channel by its total and write [NBINS, 3] layout.
// ---------------------------------------------------------------------------
__global__ __launch_bounds__(NBINS) void NH_norm(const unsigned int* __restrict__ ws,
                                                 float* __restrict__ out) {
    __shared__ float inv_tot[NCH];
    if (threadIdx.x < NCH) {
        unsigned long long t = 0;
        for (int b = 0; b < NBINS; ++b) t += ws[threadIdx.x * NBINS + b];
        inv_tot[threadIdx.x] = (t != 0) ? (1.0f / (float)t) : 0.0f;
    }
    __syncthreads();
    const int b = threadIdx.x;
#pragma unroll
    for (int c = 0; c < NCH; ++c) {
        out[b * NCH + c] = (float)ws[c * NBINS + b] * inv_tot[c];
    }
}

// ---------------------------------------------------------------------------
extern "C" void kernel_launch(void* const* d_in, const int* in_sizes, int n_in,
                              void* d_out, int out_size, void* d_ws, size_t ws_size,
                              hipStream_t stream) {
    const float* in  = (const float*)d_in[0];
    unsigned int* ws = (unsigned int*)d_ws;   // 768 * 4 B = 3 KB scratch
    float* out       = (float*)d_out;         // 768 floats, [NBINS, 3]

    const long long n  = (long long)in_sizes[0];  // 64*512*512*3
    const long long n4 = n / 4;                   // exactly divisible

    NH_zero<<<(HSIZE + BLOCK - 1) / BLOCK, BLOCK, 0, stream>>>(ws);
    NH_hist<<<GRID, BLOCK, 0, stream>>>(in, ws, n4);
    NH_norm<<<1, NBINS, 0, stream>>>(ws, out);
}